// AdvancedMixConsole_3977139716827
// MI455X (gfx1250) — compile-verified
//
#include <hip/hip_runtime.h>
#include <hip/hip_bf16.h>
#include <math.h>

// ---------------- types for WMMA ----------------
typedef __attribute__((ext_vector_type(2))) float v2f;
typedef __attribute__((ext_vector_type(8))) float v8f;

#define NCH    32          // 4 batches * 8 tracks
#define NS     131072      // samples per channel
#define CHUNK  128         // samples per thread in biquad scan
#define BQT    1024        // threads per biquad block (BQT*CHUNK == NS)
#define CSTR   40          // floats of coeff storage per channel

static __device__ __forceinline__ float exp10_(float x) { return exp2f(x * 3.321928095f); }

// normalized biquad coeffs -> out[5] = {b0,b1,b2,a1,a2}
static __device__ void shelf_coeffs(float g, float f, float q, float sgn, float* o) {
    float A  = exp10_(g * (1.0f / 40.0f));
    float w0 = 6.283185307f * f * (1.0f / 44100.0f);
    float cw = cosf(w0), sw = sinf(w0);
    float al = sw / (2.0f * q);
    float s2 = 2.0f * sqrtf(A) * al;
    float b0 = A * ((A + 1.0f) - sgn * (A - 1.0f) * cw + s2);
    float b1 = sgn * 2.0f * A * ((A - 1.0f) - sgn * (A + 1.0f) * cw);
    float b2 = A * ((A + 1.0f) - sgn * (A - 1.0f) * cw - s2);
    float a0 = (A + 1.0f) + sgn * (A - 1.0f) * cw + s2;
    float a1 = -sgn * 2.0f * ((A - 1.0f) + sgn * (A + 1.0f) * cw);
    float a2 = (A + 1.0f) + sgn * (A - 1.0f) * cw - s2;
    float r  = 1.0f / a0;
    o[0] = b0 * r; o[1] = b1 * r; o[2] = b2 * r; o[3] = a1 * r; o[4] = a2 * r;
}

static __device__ void peak_coeffs(float g, float f, float q, float* o) {
    float A  = exp10_(g * (1.0f / 40.0f));
    float w0 = 6.283185307f * f * (1.0f / 44100.0f);
    float cw = cosf(w0), sw = sinf(w0);
    float al = sw / (2.0f * q);
    float b0 = 1.0f + al * A, b1 = -2.0f * cw, b2 = 1.0f - al * A;
    float a0 = 1.0f + al / A, a1 = -2.0f * cw, a2 = 1.0f - al / A;
    float r  = 1.0f / a0;
    o[0] = b0 * r; o[1] = b1 * r; o[2] = b2 * r; o[3] = a1 * r; o[4] = a2 * r;
}

// ------------- kernel 1: per-channel parameter denorm + coefficients -------------
__global__ void setup_kernel(const float* __restrict__ params, float* __restrict__ coef) {
    int c = threadIdx.x;
    if (c >= NCH) return;
    const float lo[26] = {-24.f,-20.f,20.f,0.1f,-20.f,80.f,0.1f,-20.f,200.f,0.1f,-20.f,500.f,0.1f,
                          -20.f,1000.f,0.1f,-20.f,4000.f,0.1f,-60.f,1.f,1.f,10.f,0.1f,0.f,0.f};
    const float hi[26] = { 24.f, 20.f,2000.f,6.f, 20.f,2000.f,6.f, 20.f,4000.f,6.f, 20.f,8000.f,6.f,
                           20.f,12000.f,6.f, 20.f,18000.f,6.f, 0.f,10.f,100.f,1000.f,24.f,12.f,1.f};
    float p[26];
    #pragma unroll
    for (int i = 0; i < 26; ++i) p[i] = params[c * 26 + i] * (hi[i] - lo[i]) + lo[i];
    float* cf = coef + c * CSTR;
    cf[0] = exp10_(p[0] * (1.0f / 20.0f));                 // input gain (linear)
    shelf_coeffs(p[1],  p[2],  p[3],  1.0f, cf + 1);       // low shelf
    peak_coeffs (p[4],  p[5],  p[6],        cf + 6);
    peak_coeffs (p[7],  p[8],  p[9],        cf + 11);
    peak_coeffs (p[10], p[11], p[12],       cf + 16);
    peak_coeffs (p[13], p[14], p[15],       cf + 21);
    shelf_coeffs(p[16], p[17], p[18], -1.0f, cf + 26);     // high shelf
    cf[31] = p[19];                                        // threshold (dB)
    cf[32] = 1.0f / p[20];                                 // 1/ratio
    cf[33] = expf(-1000.0f / (44100.0f * p[21]));          // attack alpha
    cf[34] = expf(-1000.0f / (44100.0f * p[22]));          // release alpha
    cf[35] = p[23];                                        // knee
    cf[36] = p[24];                                        // makeup (dB)
    float th = p[25] * 1.570796327f;
    cf[37] = cosf(th);                                     // pan L
    cf[38] = sinf(th);                                     // pan R
}

// ------------- kernel 2: 6 cascaded biquads as parallel linear scans -------------
// One workgroup per channel. Chunked scan: forced pass -> boundary-state
// propagation with A^CHUNK -> homogeneous correction (y += [A^j S]_0).
__global__ __launch_bounds__(BQT) void biquad_kernel(const float* __restrict__ tracks,
                                                     const float* __restrict__ coef,
                                                     float* __restrict__ xbuf) {
    __shared__ float F1[BQT], F2[BQT], S1[BQT], S2[BQT];
    const int c = blockIdx.x;
    const int t = threadIdx.x;
    const float* cf = coef + c * CSTR;
    const float gain = cf[0];
    float* xc = xbuf + (size_t)c * NS;
    const float* tr = tracks + (size_t)c * NS;
    const int base = t * CHUNK;

    for (int st = 0; st < 6; ++st) {
        const float b0 = cf[1 + st * 5 + 0];
        const float b1 = cf[1 + st * 5 + 1];
        const float b2 = cf[1 + st * 5 + 2];
        const float a1 = cf[1 + st * 5 + 3];
        const float a2 = cf[1 + st * 5 + 4];

        // pass 1: forced response within chunk (zero initial state)
        float s1 = 0.f, s2 = 0.f;
        for (int j = 0; j < CHUNK; j += 4) {
            float4 xi;
            if (st == 0) {
                xi = *(const float4*)(tr + base + j);
                xi.x *= gain; xi.y *= gain; xi.z *= gain; xi.w *= gain;
            } else {
                xi = *(const float4*)(xc + base + j);
            }
            float4 yo;
            { float y = b0 * xi.x + s1; float n1 = b1 * xi.x - a1 * y + s2; s2 = b2 * xi.x - a2 * y; s1 = n1; yo.x = y; }
            { float y = b0 * xi.y + s1; float n1 = b1 * xi.y - a1 * y + s2; s2 = b2 * xi.y - a2 * y; s1 = n1; yo.y = y; }
            { float y = b0 * xi.z + s1; float n1 = b1 * xi.z - a1 * y + s2; s2 = b2 * xi.z - a2 * y; s1 = n1; yo.z = y; }
            { float y = b0 * xi.w + s1; float n1 = b1 * xi.w - a1 * y + s2; s2 = b2 * xi.w - a2 * y; s1 = n1; yo.w = y; }
            *(float4*)(xc + base + j) = yo;
        }
        F1[t] = s1; F2[t] = s2;
        __syncthreads();

        // boundary-state scan: S_{i+1} = A^CHUNK * S_i + F_i   (A = [[-a1,1],[-a2,0]])
        if (t == 0) {
            float m00 = -a1, m01 = 1.f, m10 = -a2, m11 = 0.f;
            #pragma unroll
            for (int k = 0; k < 7; ++k) {      // A^(2^7) = A^128 = A^CHUNK
                float n00 = m00 * m00 + m01 * m10;
                float n01 = m00 * m01 + m01 * m11;
                float n10 = m10 * m00 + m11 * m10;
                float n11 = m10 * m01 + m11 * m11;
                m00 = n00; m01 = n01; m10 = n10; m11 = n11;
            }
            float u1 = 0.f, u2 = 0.f;
            for (int i = 0; i < BQT; ++i) {
                S1[i] = u1; S2[i] = u2;
                float f1 = F1[i], f2 = F2[i];
                float n1 = m00 * u1 + m01 * u2 + f1;
                float n2 = m10 * u1 + m11 * u2 + f2;
                u1 = n1; u2 = n2;
            }
        }
        __syncthreads();

        // pass 2: add homogeneous part, v <- A v each sample, y += v0
        float v0 = S1[t], v1 = S2[t];
        for (int j = 0; j < CHUNK; j += 4) {
            float4 y = *(float4*)(xc + base + j);
            y.x += v0; { float n0 = -a1 * v0 + v1; v1 = -a2 * v0; v0 = n0; }
            y.y += v0; { float n0 = -a1 * v0 + v1; v1 = -a2 * v0; v0 = n0; }
            y.z += v0; { float n0 = -a1 * v0 + v1; v1 = -a2 * v0; v0 = n0; }
            y.w += v0; { float n0 = -a1 * v0 + v1; v1 = -a2 * v0; v0 = n0; }
            *(float4*)(xc + base + j) = y;
        }
        __syncthreads();
    }
}

// ------------- kernel 3: compressor static gain computer (elementwise) -------------
__global__ void gc_kernel(const float* __restrict__ xbuf, const float* __restrict__ coef,
                          float* __restrict__ gcbuf) {
    int c = blockIdx.y;
    size_t i = (size_t)c * NS + blockIdx.x * blockDim.x + threadIdx.x;
    float thr  = coef[c * CSTR + 31];
    float rinv = coef[c * CSTR + 32];
    float knee = coef[c * CSTR + 35];
    float x   = xbuf[i];
    float xdb = 6.020599913f * __log2f(fmaxf(fabsf(x), 1e-8f)); // 20*log10
    float d   = xdb - thr;
    float gc;
    if (2.0f * d < -knee)      gc = 0.0f;
    else if (2.0f * d > knee)  gc = d * (rinv - 1.0f);
    else { float u = d + 0.5f * knee; gc = (rinv - 1.0f) * u * u / (2.0f * knee); }
    gcbuf[i] = gc;
}

// ------------- kernel 4: data-dependent envelope (serial in time, lane=channel) -------------
__global__ void smooth_kernel(float* __restrict__ gcbuf, const float* __restrict__ coef) {
    int c = threadIdx.x;
    if (c >= NCH) return;
    float aa = coef[c * CSTR + 33];
    float ar = coef[c * CSTR + 34];
    float* g = gcbuf + (size_t)c * NS;
    float e = 0.f;
    for (int t = 0; t < NS; ++t) {
        float gc = g[t];
        float al = (gc < e) ? aa : ar;
        e = al * e + (1.0f - al) * gc;
        g[t] = e;
    }
}

// ------------- kernel 5: apply smoothed gain + makeup (elementwise) -------------
__global__ void apply_kernel(float* __restrict__ xbuf, const float* __restrict__ coef,
                             const float* __restrict__ gcbuf) {
    int c = blockIdx.y;
    size_t i = (size_t)c * NS + blockIdx.x * blockDim.x + threadIdx.x;
    float mk = coef[c * CSTR + 36];
    xbuf[i] = xbuf[i] * exp2f((gcbuf[i] + mk) * 0.166096404744f);   // 10^(dB/20)
}

// ------------- kernel 6: pan + mixdown = (16x32)x(32x16) GEMM tiles on WMMA -------------
// G[r][k]: r=0..7 real rows (batch=r>>1, L/R=r&1), nonzero iff k>>3 == r>>1.
static __device__ __forceinline__ float gval(const float* __restrict__ coef, int M, int K) {
    if (M >= 8 || (K >> 3) != (M >> 1)) return 0.0f;
    return (M & 1) ? coef[K * CSTR + 38] : coef[K * CSTR + 37];
}

__global__ void mix_kernel(const float* __restrict__ xbuf, const float* __restrict__ coef,
                           float* __restrict__ out) {
    const int lane = threadIdx.x & 31;
    const int wave = (int)((blockIdx.x * blockDim.x + threadIdx.x) >> 5);
    const int n0   = wave * 16;                  // 16-sample tile per wave
    const int M    = lane & 15;
    const int koff = (lane < 16) ? 0 : 2;        // A/B fragment K offset (ISA 16x4 layout)
    const int nn   = n0 + (lane & 15);

    // A fragments (pan-gain matrix), 8 chunks of K=4
    v2f afrag[8];
    #pragma unroll
    for (int kc = 0; kc < 8; ++kc) {
        int k = kc * 4 + koff;
        afrag[kc].x = gval(coef, M, k);
        afrag[kc].y = gval(coef, M, k + 1);
    }

    v8f acc = {};
    #pragma unroll
    for (int kc = 0; kc < 8; ++kc) {
        int k = kc * 4 + koff;
        v2f b;
        b.x = xbuf[(size_t)k * NS + nn];
        b.y = xbuf[(size_t)(k + 1) * NS + nn];
        acc = __builtin_amdgcn_wmma_f32_16x16x4_f32(false, afrag[kc], false, b,
                                                    (short)0, acc, false, false);
    }

    // D layout: VGPR v -> row v (lanes 0-15), row v+8 (lanes 16-31, padding -> discard)
    if (lane < 16) {
        #pragma unroll
        for (int v = 0; v < 8; ++v) out[(size_t)v * NS + n0 + lane] = acc[v];
    }
}

extern "C" void kernel_launch(void* const* d_in, const int* in_sizes, int n_in,
                              void* d_out, int out_size, void* d_ws, size_t ws_size,
                              hipStream_t stream) {
    const float* tracks = (const float*)d_in[0];   // (4,8,131072) f32
    const float* params = (const float*)d_in[1];   // (4,8,26) f32
    float* out  = (float*)d_out;                   // (4,2,131072) f32

    float* coef  = (float*)d_ws;                               // 32*40 floats
    float* xbuf  = (float*)((char*)d_ws + 8192);               // 32*131072 f32 = 16 MB
    float* gcbuf = xbuf + (size_t)NCH * NS;                    // 16 MB

    setup_kernel <<<1, 32, 0, stream>>>(params, coef);
    biquad_kernel<<<NCH, BQT, 0, stream>>>(tracks, coef, xbuf);
    dim3 egrid(NS / 256, NCH);
    gc_kernel    <<<egrid, 256, 0, stream>>>(xbuf, coef, gcbuf);
    smooth_kernel<<<1, 32, 0, stream>>>(gcbuf, coef);
    apply_kernel <<<egrid, 256, 0, stream>>>(xbuf, coef, gcbuf);
    // NS/16 tiles, 8 waves (256 threads) per block -> exact grid, full EXEC at WMMA
    mix_kernel   <<<(NS / 16) / 8, 256, 0, stream>>>(xbuf, coef, out);
    (void)in_sizes; (void)n_in; (void)out_size; (void)ws_size;
}